// CompGCNCov_89799176224927
// MI455X (gfx1250) — compile-verified
//
#include <hip/hip_runtime.h>

#define D       200
#define NNODES  100000
#define NEDGES  400000
#define HALF_BLOCKS 12500   // 200000 edges / 16 per block
#define KSTEPS  50          // 200 / 4
#define ETILE   16
#define NTILES  13          // ceil(200/16)
#define BLOCK   (NTILES * 32)   // 416 threads = 13 waves, one N-tile per wave
#define STRIDE  201         // LDS row pad: (m*201*4B) -> distinct banks for m=0..15
#define BN_EPS  1e-5f
#define ONE_THIRD 0.33333333333333333f

typedef float v2f __attribute__((ext_vector_type(2)));
typedef float v8f __attribute__((ext_vector_type(8)));

// ---------------------------------------------------------------------------
// Zero agg region of d_out + BN stats in ws
// ---------------------------------------------------------------------------
__global__ __launch_bounds__(256) void zero_kernel(float* __restrict__ h, int n,
                                                   float* __restrict__ stats) {
    int idx = blockIdx.x * blockDim.x + threadIdx.x;
    int stride = gridDim.x * blockDim.x;
    for (int i = idx; i < n; i += stride) h[i] = 0.0f;
    if (idx < 2 * D) stats[idx] = 0.0f;
}

// ---------------------------------------------------------------------------
// One 16x16 output tile of ed[16 x 200] @ W[200 x 200] per wave.
// n must be pre-clamped to < D (garbage columns are never stored by caller).
// Two accumulator chains (even/odd K-step) + next-step prefetch so loads for
// step k+2 are in flight while the WMMAs for step k issue.
// ---------------------------------------------------------------------------
__device__ __forceinline__ v8f wmma_tile(const float* __restrict__ ed_lds,
                                         const float* __restrict__ W,
                                         int lane, int nclamped) {
    const int mrow  = lane & 15;
    const int khalf = (lane >> 4) * 2;

    const float* __restrict__ ap = ed_lds + mrow * STRIDE + khalf;
    const float* __restrict__ wp = W + khalf * D + nclamped;

    v8f accA = {}, accB = {};
    v2f aA, bA, aB, bB;
    aA.x = ap[0];     aA.y = ap[1];
    bA.x = wp[0];     bA.y = wp[D];
    aB.x = ap[4];     aB.y = ap[5];
    bB.x = wp[4 * D]; bB.y = wp[5 * D];

    for (int k = 0; k < KSTEPS - 2; k += 2) {
        ap += 8;
        wp += 8 * D;
        v2f naA, nbA, naB, nbB;
        naA.x = ap[0];     naA.y = ap[1];
        nbA.x = wp[0];     nbA.y = wp[D];
        naB.x = ap[4];     naB.y = ap[5];
        nbB.x = wp[4 * D]; nbB.y = wp[5 * D];
        accA = __builtin_amdgcn_wmma_f32_16x16x4_f32(
            false, aA, false, bA, (short)0, accA, false, false);
        accB = __builtin_amdgcn_wmma_f32_16x16x4_f32(
            false, aB, false, bB, (short)0, accB, false, false);
        aA = naA; bA = nbA; aB = naB; bB = nbB;
    }
    accA = __builtin_amdgcn_wmma_f32_16x16x4_f32(
        false, aA, false, bA, (short)0, accA, false, false);
    accB = __builtin_amdgcn_wmma_f32_16x16x4_f32(
        false, aB, false, bB, (short)0, accB, false, false);
    return accA + accB;
}

// ---------------------------------------------------------------------------
// Edge messages: msg = (x[src] * rel[type]) @ W, scaled by edge_norm,
// atomically scattered into agg[dst]. 16 edges per block, 13 waves,
// one 16-wide N-tile per wave (no inter-wave imbalance, no branches).
// ---------------------------------------------------------------------------
__global__ __launch_bounds__(BLOCK) void edge_msg_kernel(
    const float* __restrict__ x, const float* __restrict__ rel,
    const float* __restrict__ edge_norm,
    const float* __restrict__ in_w, const float* __restrict__ out_w,
    const int* __restrict__ edge_src, const int* __restrict__ edge_dst,
    const int* __restrict__ edge_type, float* __restrict__ agg) {
    __shared__ float ed[ETILE * STRIDE];
    __shared__ int   s_src[ETILE], s_type[ETILE], s_dst[ETILE];
    __shared__ float s_norm[ETILE];

    const int tid = threadIdx.x;
    const int e0  = blockIdx.x * ETILE;
    const float* __restrict__ W = (blockIdx.x < HALF_BLOCKS) ? in_w : out_w;

    if (tid < ETILE) {
        s_src[tid]  = edge_src[e0 + tid];
        s_type[tid] = edge_type[e0 + tid];
        s_dst[tid]  = edge_dst[e0 + tid];
        s_norm[tid] = edge_norm[e0 + tid];
    }
    __syncthreads();

    // Stage edge_data[16][200] = x[src] * rel[type] into LDS
    for (int idx = tid; idx < ETILE * D; idx += BLOCK) {
        int i = idx / D, d = idx - i * D;
        ed[i * STRIDE + d] = x[s_src[i] * D + d] * rel[s_type[i] * D + d];
    }
    __syncthreads();

    const int wave = tid >> 5;            // 0..12 == N-tile index
    const int lane = tid & 31;
    const int mrow = lane & 15;
    const int n    = wave * 16 + mrow;    // up to 207
    const int nc   = (n < D) ? n : (D - 1);

    const v8f acc = wmma_tile(ed, W, lane, nc);

    // D layout: VGPR r -> M=r (lanes 0-15) / M=r+8 (lanes 16-31)
    const int mofs = (lane >> 4) << 3;
    const bool ok = (n < D);
    for (int r = 0; r < 8; ++r) {
        const int m = r + mofs;
        if (ok) atomicAdd(&agg[s_dst[m] * D + n], acc[r] * s_norm[m]);
    }
}

// ---------------------------------------------------------------------------
// Self-loop term: h = agg/3 + ((x * loop_rel) @ loop_w)/3, in place in d_out.
// ---------------------------------------------------------------------------
__global__ __launch_bounds__(BLOCK) void node_loop_kernel(
    const float* __restrict__ x, const float* __restrict__ loop_rel,
    const float* __restrict__ loop_w, float* __restrict__ h) {
    __shared__ float ed[ETILE * STRIDE];
    const int tid = threadIdx.x;
    const int node0 = blockIdx.x * ETILE;

    for (int idx = tid; idx < ETILE * D; idx += BLOCK) {
        int i = idx / D, d = idx - i * D;
        ed[i * STRIDE + d] = x[(node0 + i) * D + d] * loop_rel[d];
    }
    __syncthreads();

    const int wave = tid >> 5;
    const int lane = tid & 31;
    const int mrow = lane & 15;
    const int n    = wave * 16 + mrow;
    const int nc   = (n < D) ? n : (D - 1);

    const v8f acc = wmma_tile(ed, loop_w, lane, nc);

    const int mofs = (lane >> 4) << 3;
    const bool ok = (n < D);
    for (int r = 0; r < 8; ++r) {
        const int row = (node0 + r + mofs) * D;
        if (ok) h[row + n] = h[row + n] * ONE_THIRD + acc[r] * ONE_THIRD;
    }
}

// ---------------------------------------------------------------------------
// Per-column sum / sumsq over 100k rows. 200 blocks x 500 rows.
// ---------------------------------------------------------------------------
__global__ __launch_bounds__(256) void stats_kernel(const float* __restrict__ h,
                                                    float* __restrict__ stats) {
    const int c = threadIdx.x;
    if (c >= D) return;
    const int r0 = blockIdx.x * 500;
    float s = 0.0f, q = 0.0f;
    for (int r = 0; r < 500; ++r) {
        const float v = h[(r0 + r) * D + c];
        s += v;
        q += v * v;
    }
    atomicAdd(&stats[c], s);
    atomicAdd(&stats[D + c], q);
}

// Fold mean/var/gamma/beta into per-column scale+bias
__global__ __launch_bounds__(256) void bn_finalize_kernel(
    const float* __restrict__ stats, const float* __restrict__ gamma,
    const float* __restrict__ beta, float* __restrict__ coef) {
    const int c = blockIdx.x * blockDim.x + threadIdx.x;
    if (c < D) {
        const float invN = 1.0f / (float)NNODES;
        const float mean = stats[c] * invN;
        const float var  = stats[D + c] * invN - mean * mean;
        const float sc   = gamma[c] * rsqrtf(var + BN_EPS);
        coef[c]     = sc;
        coef[D + c] = beta[c] - mean * sc;
    }
}

__global__ __launch_bounds__(256) void bn_apply_kernel(float* __restrict__ h,
                                                       const float* __restrict__ coef,
                                                       int n) {
    int idx = blockIdx.x * blockDim.x + threadIdx.x;
    const int stride = gridDim.x * blockDim.x;
    for (int i = idx; i < n; i += stride) {
        const int c = i % D;
        h[i] = h[i] * coef[c] + coef[D + c];
    }
}

// ---------------------------------------------------------------------------
// rel_out = rel_repr @ w_rel  (200x200x200), one wave per 16x16 tile.
// Clamped A rows / B cols; dual accumulator chains with prefetch.
// ---------------------------------------------------------------------------
__global__ __launch_bounds__(32) void rel_out_kernel(
    const float* __restrict__ rel, const float* __restrict__ w_rel,
    float* __restrict__ out) {
    const int tn = blockIdx.x, tm = blockIdx.y;   // 13 x 13 tiles
    const int lane  = threadIdx.x;
    const int mrow  = lane & 15;
    const int khalf = (lane >> 4) * 2;
    const int m = tm * 16 + mrow;                 // A row, up to 207
    const int n = tn * 16 + mrow;                 // B col, up to 207
    const int mc = (m < D) ? m : (D - 1);
    const int nc = (n < D) ? n : (D - 1);

    const float* __restrict__ ap = rel + mc * D + khalf;
    const float* __restrict__ wp = w_rel + khalf * D + nc;

    v8f accA = {}, accB = {};
    v2f aA, bA, aB, bB;
    aA.x = ap[0];     aA.y = ap[1];
    bA.x = wp[0];     bA.y = wp[D];
    aB.x = ap[4];     aB.y = ap[5];
    bB.x = wp[4 * D]; bB.y = wp[5 * D];

    for (int k = 0; k < KSTEPS - 2; k += 2) {
        ap += 8;
        wp += 8 * D;
        v2f naA, nbA, naB, nbB;
        naA.x = ap[0];     naA.y = ap[1];
        nbA.x = wp[0];     nbA.y = wp[D];
        naB.x = ap[4];     naB.y = ap[5];
        nbB.x = wp[4 * D]; nbB.y = wp[5 * D];
        accA = __builtin_amdgcn_wmma_f32_16x16x4_f32(
            false, aA, false, bA, (short)0, accA, false, false);
        accB = __builtin_amdgcn_wmma_f32_16x16x4_f32(
            false, aB, false, bB, (short)0, accB, false, false);
        aA = naA; bA = nbA; aB = naB; bB = nbB;
    }
    accA = __builtin_amdgcn_wmma_f32_16x16x4_f32(
        false, aA, false, bA, (short)0, accA, false, false);
    accB = __builtin_amdgcn_wmma_f32_16x16x4_f32(
        false, aB, false, bB, (short)0, accB, false, false);
    const v8f acc = accA + accB;

    const int mofs = (lane >> 4) << 3;
    for (int r = 0; r < 8; ++r) {
        const int mm = tm * 16 + r + mofs;
        const int nn = tn * 16 + (lane & 15);
        if (mm < D && nn < D) out[mm * D + nn] = acc[r];
    }
}

// ---------------------------------------------------------------------------
extern "C" void kernel_launch(void* const* d_in, const int* in_sizes, int n_in,
                              void* d_out, int out_size, void* d_ws, size_t ws_size,
                              hipStream_t stream) {
    (void)in_sizes; (void)n_in; (void)out_size; (void)ws_size;
    const float* x         = (const float*)d_in[0];
    const float* rel       = (const float*)d_in[1];
    const float* edge_norm = (const float*)d_in[2];
    const float* in_w      = (const float*)d_in[3];
    const float* out_w     = (const float*)d_in[4];
    const float* loop_w    = (const float*)d_in[5];
    const float* loop_rel  = (const float*)d_in[6];
    const float* w_rel     = (const float*)d_in[7];
    const float* gamma     = (const float*)d_in[8];
    const float* beta      = (const float*)d_in[9];
    const int*   edge_src  = (const int*)d_in[10];
    const int*   edge_dst  = (const int*)d_in[11];
    const int*   edge_type = (const int*)d_in[12];

    float* h       = (float*)d_out;                   // [NNODES * D], doubles as agg
    float* rel_out = (float*)d_out + NNODES * D;      // [D * D]
    float* stats   = (float*)d_ws;                    // [2*D] sums, then [2*D] coef
    float* coef    = stats + 2 * D;

    const int NH = NNODES * D;

    zero_kernel<<<4096, 256, 0, stream>>>(h, NH, stats);
    edge_msg_kernel<<<NEDGES / ETILE, BLOCK, 0, stream>>>(
        x, rel, edge_norm, in_w, out_w, edge_src, edge_dst, edge_type, h);
    node_loop_kernel<<<NNODES / ETILE, BLOCK, 0, stream>>>(x, loop_rel, loop_w, h);
    stats_kernel<<<200, 256, 0, stream>>>(h, stats);
    bn_finalize_kernel<<<1, 256, 0, stream>>>(stats, gamma, beta, coef);
    bn_apply_kernel<<<1024, 256, 0, stream>>>(h, coef, NH);
    rel_out_kernel<<<dim3(13, 13), 32, 0, stream>>>(rel, w_rel, rel_out);
}